// SMPL_8246337208992
// MI455X (gfx1250) — compile-verified
//
#include <hip/hip_runtime.h>
#include <hip/hip_bf16.h>

typedef __attribute__((ext_vector_type(2))) float v2f;
typedef __attribute__((ext_vector_type(8))) float v8f;

#define SMPL_B  512
#define SMPL_V  6890
#define SMPL_M  (SMPL_V * 3)   // 20670 rows of the vertex GEMMs
#define SMPL_NJ 24
#define TN      4              // N-tiles per wave
#define TM      2              // M-tiles per wave (32x64 strip per wave)

// ---------------------------------------------------------------------------
// WMMA GEMM for blendshape application:
//   io[n*M + m] = init + sum_k A[m*K+k] * B[k][n]
//   A: (M x K) row-major.  BTRANS: B[k][n] = Bsrc[n*K+k] else Bsrc[k*N+n]
//   INITMODE 0: C0 = tmpl[m]      (v_shaped)
//   INITMODE 1: C0 = io[n*M+m]    (v_posed, in-place accumulate)
// One wave computes a 32x64 strip: 2 A frags x 4 B frags -> 8 WMMAs / k-step.
// ---------------------------------------------------------------------------
template <int BTRANS, int INITMODE>
__global__ void k_wmma_dirs(const float* __restrict__ A, int K,
                            const float* __restrict__ Bsrc,
                            const float* __restrict__ tmpl,
                            float* __restrict__ io, int M, int N)
{
    const int lane = threadIdx.x & 31;
    const int wave = threadIdx.x >> 5;
    const int row  = lane & 15;
    const int hi   = lane >> 4;
    const int m0   = (blockIdx.x * 4 + wave) * (16 * TM);
    const int n0   = blockIdx.y * (16 * TN);
    const int nl   = n0 + row;                   // lane's base column

    const float* arow[TM];
    #pragma unroll
    for (int u = 0; u < TM; ++u) {
        int m  = m0 + 16 * u + row;
        int mc = (m < M) ? m : (M - 1);          // clamped: loads always legal
        arow[u] = A + (size_t)mc * K;
    }

    v8f c[TM][TN];
    #pragma unroll
    for (int u = 0; u < TM; ++u)
        #pragma unroll
        for (int t = 0; t < TN; ++t)
            #pragma unroll
            for (int r = 0; r < 8; ++r) {
                int mr  = m0 + 16 * u + r + hi * 8;
                int mrc = (mr < M) ? mr : (M - 1);
                c[u][t][r] = (INITMODE == 0)
                               ? tmpl[mrc]
                               : io[(size_t)(nl + 16 * t) * M + mrc];
            }

    const int Kmain = K & ~3;

    #pragma unroll 2
    for (int kk = 0; kk < Kmain; kk += 4) {
        const int ka = kk + 2 * hi;
        v2f a[TM], b[TN];
        #pragma unroll
        for (int u = 0; u < TM; ++u) {
            a[u].x = arow[u][ka];
            a[u].y = arow[u][ka + 1];
        }
        #pragma unroll
        for (int t = 0; t < TN; ++t) {
            if (BTRANS) {
                const float* bp = Bsrc + (size_t)(nl + 16 * t) * K;
                b[t].x = bp[ka];
                b[t].y = bp[ka + 1];
            } else {
                b[t].x = Bsrc[(size_t)ka * N + nl + 16 * t];
                b[t].y = Bsrc[(size_t)(ka + 1) * N + nl + 16 * t];
            }
        }
        #pragma unroll
        for (int u = 0; u < TM; ++u)
            #pragma unroll
            for (int t = 0; t < TN; ++t)
                c[u][t] = __builtin_amdgcn_wmma_f32_16x16x4_f32(
                    false, a[u], false, b[t], (short)0, c[u][t], false, false);
    }

    if (Kmain < K) {                             // one guarded tail step
        const int ka = Kmain + 2 * hi;
        v2f a[TM], b[TN];
        #pragma unroll
        for (int u = 0; u < TM; ++u) {
            a[u].x = (ka     < K) ? arow[u][ka]     : 0.0f;
            a[u].y = (ka + 1 < K) ? arow[u][ka + 1] : 0.0f;
        }
        #pragma unroll
        for (int t = 0; t < TN; ++t) {
            if (BTRANS) {
                const float* bp = Bsrc + (size_t)(nl + 16 * t) * K;
                b[t].x = (ka     < K) ? bp[ka]     : 0.0f;
                b[t].y = (ka + 1 < K) ? bp[ka + 1] : 0.0f;
            } else {
                b[t].x = (ka     < K) ? Bsrc[(size_t)ka * N + nl + 16 * t]       : 0.0f;
                b[t].y = (ka + 1 < K) ? Bsrc[(size_t)(ka + 1) * N + nl + 16 * t] : 0.0f;
            }
        }
        #pragma unroll
        for (int u = 0; u < TM; ++u)
            #pragma unroll
            for (int t = 0; t < TN; ++t)
                c[u][t] = __builtin_amdgcn_wmma_f32_16x16x4_f32(
                    false, a[u], false, b[t], (short)0, c[u][t], false, false);
    }

    #pragma unroll
    for (int u = 0; u < TM; ++u)
        #pragma unroll
        for (int t = 0; t < TN; ++t)
            #pragma unroll
            for (int r = 0; r < 8; ++r) {
                int mr = m0 + 16 * u + r + hi * 8;
                if (mr < M) io[(size_t)(nl + 16 * t) * M + mr] = c[u][t][r];
            }
}

// ---------------------------------------------------------------------------
// WMMA GEMM for joint regression:
//   C (Mj x N=B*3) = Areg(Mj x V) @ Vm,  Vm[v][n] = vsrc[(n/3)*SMPL_M + v*3 + n%3]
//   A rows from up to 3 concatenated regressors (s1, s2, rest).
//   out[(n/3)*(Mj*3) + m*3 + (n%3)] = C[m][n]
// One wave covers 32 rows; 2 waves per block cover Mj<=64.
// ---------------------------------------------------------------------------
__global__ void k_wmma_reg(const float* __restrict__ r1, int s1,
                           const float* __restrict__ r2, int s2,
                           const float* __restrict__ r3, int Mj,
                           const float* __restrict__ vsrc,
                           float* __restrict__ out, int N)
{
    const int lane = threadIdx.x & 31;
    const int wave = threadIdx.x >> 5;
    const int row  = lane & 15;
    const int hi   = lane >> 4;
    const int m0   = (blockIdx.x * 2 + wave) * (16 * TM);
    const int n0   = blockIdx.y * (16 * TN);
    const int nl   = n0 + row;

    const float* arow[TM];
    #pragma unroll
    for (int u = 0; u < TM; ++u) {
        int m    = m0 + 16 * u + row;
        int meff = (m < Mj) ? m : 0;             // clamp: loads always legal
        if (meff < s1)            arow[u] = r1 + (size_t)meff * SMPL_V;
        else if (meff < s1 + s2)  arow[u] = r2 + (size_t)(meff - s1) * SMPL_V;
        else                      arow[u] = r3 + (size_t)(meff - s1 - s2) * SMPL_V;
    }

    const float* bptr[TN];
    #pragma unroll
    for (int t = 0; t < TN; ++t) {
        int n = nl + 16 * t;
        bptr[t] = vsrc + (size_t)(n / 3) * SMPL_M + (n % 3);
    }

    v8f c[TM][TN];
    #pragma unroll
    for (int u = 0; u < TM; ++u)
        #pragma unroll
        for (int t = 0; t < TN; ++t)
            #pragma unroll
            for (int r = 0; r < 8; ++r) c[u][t][r] = 0.0f;

    const int Kmain = SMPL_V & ~3;               // 6888

    #pragma unroll 2
    for (int kk = 0; kk < Kmain; kk += 4) {
        const int ka = kk + 2 * hi;
        v2f a[TM], b[TN];
        #pragma unroll
        for (int u = 0; u < TM; ++u) {
            a[u].x = arow[u][ka];
            a[u].y = arow[u][ka + 1];
        }
        #pragma unroll
        for (int t = 0; t < TN; ++t) {
            b[t].x = bptr[t][(size_t)ka * 3];
            b[t].y = bptr[t][(size_t)(ka + 1) * 3];
        }
        #pragma unroll
        for (int u = 0; u < TM; ++u)
            #pragma unroll
            for (int t = 0; t < TN; ++t)
                c[u][t] = __builtin_amdgcn_wmma_f32_16x16x4_f32(
                    false, a[u], false, b[t], (short)0, c[u][t], false, false);
    }

    {   // tail: SMPL_V % 4 == 2
        const int ka = Kmain + 2 * hi;
        v2f a[TM], b[TN];
        #pragma unroll
        for (int u = 0; u < TM; ++u) {
            a[u].x = (ka     < SMPL_V) ? arow[u][ka]     : 0.0f;
            a[u].y = (ka + 1 < SMPL_V) ? arow[u][ka + 1] : 0.0f;
        }
        #pragma unroll
        for (int t = 0; t < TN; ++t) {
            b[t].x = (ka     < SMPL_V) ? bptr[t][(size_t)ka * 3]       : 0.0f;
            b[t].y = (ka + 1 < SMPL_V) ? bptr[t][(size_t)(ka + 1) * 3] : 0.0f;
        }
        #pragma unroll
        for (int u = 0; u < TM; ++u)
            #pragma unroll
            for (int t = 0; t < TN; ++t)
                c[u][t] = __builtin_amdgcn_wmma_f32_16x16x4_f32(
                    false, a[u], false, b[t], (short)0, c[u][t], false, false);
    }

    #pragma unroll
    for (int t = 0; t < TN; ++t) {
        int n = nl + 16 * t;
        int nb = n / 3, nd = n % 3;
        #pragma unroll
        for (int u = 0; u < TM; ++u)
            #pragma unroll
            for (int r = 0; r < 8; ++r) {
                int mr = m0 + 16 * u + r + hi * 8;
                if (mr < Mj) out[(size_t)nb * (Mj * 3) + mr * 3 + nd] = c[u][t][r];
            }
    }
}

// ---------------------------------------------------------------------------
// Per-batch: Rodrigues, lrotmin^T (207 x B), kinematic chain, Gc (B,24,3,4).
// One thread per batch, chain matrices staged in LDS.
// ---------------------------------------------------------------------------
__global__ void k_pose(const float* __restrict__ pose,
                       const float* __restrict__ Jbuf,
                       const int* __restrict__ parent,
                       float* __restrict__ lrotT,
                       float* __restrict__ Gc, int Bn)
{
    __shared__ float Als[32][SMPL_NJ][12];
    int b = blockIdx.x * 32 + threadIdx.x;
    if (b >= Bn) return;
    float (*Ach)[12] = Als[threadIdx.x];

    float Jl[SMPL_NJ][3];
    for (int j = 0; j < SMPL_NJ; ++j)
        for (int d = 0; d < 3; ++d)
            Jl[j][d] = Jbuf[(size_t)b * SMPL_NJ * 3 + j * 3 + d];

    for (int j = 0; j < SMPL_NJ; ++j) {
        float rx = pose[(size_t)b * 72 + j * 3 + 0];
        float ry = pose[(size_t)b * 72 + j * 3 + 1];
        float rz = pose[(size_t)b * 72 + j * 3 + 2];
        float th  = sqrtf(rx * rx + ry * ry + rz * rz);
        float inv = 1.0f / fmaxf(th, 1e-8f);
        float kx = rx * inv, ky = ry * inv, kz = rz * inv;
        float Km[9] = {0.f, -kz, ky,  kz, 0.f, -kx,  -ky, kx, 0.f};
        float K2[9];
        for (int x = 0; x < 3; ++x)
            for (int y = 0; y < 3; ++y) {
                float s = 0.f;
                for (int z = 0; z < 3; ++z) s += Km[x * 3 + z] * Km[z * 3 + y];
                K2[x * 3 + y] = s;
            }
        float st = sinf(th), ct = 1.0f - cosf(th);
        float R[9];
        for (int i = 0; i < 9; ++i) R[i] = st * Km[i] + ct * K2[i];
        R[0] += 1.f; R[4] += 1.f; R[8] += 1.f;

        if (j >= 1)   // lrotmin stored transposed: (207, B)
            for (int i = 0; i < 9; ++i)
                lrotT[(size_t)((j - 1) * 9 + i) * Bn + b] =
                    R[i] - ((i % 4 == 0) ? 1.0f : 0.0f);

        float t[3];
        if (j == 0) {
            t[0] = Jl[0][0]; t[1] = Jl[0][1]; t[2] = Jl[0][2];
            for (int x = 0; x < 3; ++x) {
                for (int y = 0; y < 3; ++y) Ach[0][x * 4 + y] = R[x * 3 + y];
                Ach[0][x * 4 + 3] = t[x];
            }
        } else {
            int p = parent[j - 1];
            for (int d = 0; d < 3; ++d) t[d] = Jl[j][d] - Jl[p][d];
            for (int x = 0; x < 3; ++x) {
                for (int y = 0; y < 3; ++y)
                    Ach[j][x * 4 + y] = Ach[p][x * 4 + 0] * R[0 * 3 + y] +
                                        Ach[p][x * 4 + 1] * R[1 * 3 + y] +
                                        Ach[p][x * 4 + 2] * R[2 * 3 + y];
                Ach[j][x * 4 + 3] = Ach[p][x * 4 + 0] * t[0] +
                                    Ach[p][x * 4 + 1] * t[1] +
                                    Ach[p][x * 4 + 2] * t[2] + Ach[p][x * 4 + 3];
            }
        }
    }

    for (int j = 0; j < SMPL_NJ; ++j)
        for (int x = 0; x < 3; ++x) {
            float tc = Ach[j][x * 4 + 0] * Jl[j][0] +
                       Ach[j][x * 4 + 1] * Jl[j][1] +
                       Ach[j][x * 4 + 2] * Jl[j][2];
            size_t base = (size_t)b * 288 + j * 12 + x * 4;
            Gc[base + 0] = Ach[j][x * 4 + 0];
            Gc[base + 1] = Ach[j][x * 4 + 1];
            Gc[base + 2] = Ach[j][x * 4 + 2];
            Gc[base + 3] = Ach[j][x * 4 + 3] - tc;
        }
}

// ---------------------------------------------------------------------------
// Linear blend skinning, in place on vbuf (v_posed -> v). Gc[b] in LDS.
// ---------------------------------------------------------------------------
__global__ void k_skin(const float* __restrict__ weights,
                       const float* __restrict__ Gc,
                       float* __restrict__ vbuf)
{
    __shared__ float g[SMPL_NJ * 12];
    int b = blockIdx.y;
    for (int i = threadIdx.x; i < SMPL_NJ * 12; i += blockDim.x)
        g[i] = Gc[(size_t)b * 288 + i];
    __syncthreads();

    int vtx = blockIdx.x * blockDim.x + threadIdx.x;
    if (vtx >= SMPL_V) return;

    float T[12];
    #pragma unroll
    for (int i = 0; i < 12; ++i) T[i] = 0.f;
    for (int j = 0; j < SMPL_NJ; ++j) {
        float w = weights[(size_t)vtx * SMPL_NJ + j];
        #pragma unroll
        for (int i = 0; i < 12; ++i) T[i] += w * g[j * 12 + i];
    }
    size_t base = (size_t)b * SMPL_M + (size_t)vtx * 3;
    float p0 = vbuf[base], p1 = vbuf[base + 1], p2 = vbuf[base + 2];
    #pragma unroll
    for (int d = 0; d < 3; ++d)
        vbuf[base + d] = T[d * 4 + 0] * p0 + T[d * 4 + 1] * p1 +
                         T[d * 4 + 2] * p2 + T[d * 4 + 3];
}

// ---------------------------------------------------------------------------
// joint_map gather, pelvis, joints output. One thread per batch.
// ---------------------------------------------------------------------------
__global__ void k_map(const float* __restrict__ jcat, const int* __restrict__ jmap,
                      float* __restrict__ out_joints, float* __restrict__ out_pelvis,
                      float* __restrict__ pelvis_ws, int Bn)
{
    int b = blockIdx.x * blockDim.x + threadIdx.x;
    if (b >= Bn) return;
    int s27 = jmap[27], s28 = jmap[28];
    float pel[3];
    for (int d = 0; d < 3; ++d)
        pel[d] = 0.5f * (jcat[(size_t)b * 162 + s27 * 3 + d] +
                         jcat[(size_t)b * 162 + s28 * 3 + d]);
    for (int q = 0; q < 49; ++q) {
        int src = jmap[q];
        for (int d = 0; d < 3; ++d)
            out_joints[(size_t)b * 147 + q * 3 + d] =
                jcat[(size_t)b * 162 + src * 3 + d] - pel[d];
    }
    for (int d = 0; d < 3; ++d) {
        out_pelvis[b * 3 + d] = pel[d];
        pelvis_ws[b * 3 + d]  = pel[d];
    }
}

// out_v = vbuf - pelvis
__global__ void k_sub(const float* __restrict__ vbuf,
                      const float* __restrict__ pelvis,
                      float* __restrict__ out_v, int total)
{
    int i = blockIdx.x * blockDim.x + threadIdx.x;
    if (i >= total) return;
    int b = i / SMPL_M;
    int d = i % 3;
    out_v[i] = vbuf[i] - pelvis[b * 3 + d];
}

// ---------------------------------------------------------------------------
extern "C" void kernel_launch(void* const* d_in, const int* in_sizes, int n_in,
                              void* d_out, int out_size, void* d_ws, size_t ws_size,
                              hipStream_t stream)
{
    const float* pose      = (const float*)d_in[0];
    const float* beta      = (const float*)d_in[1];
    const float* v_tmpl    = (const float*)d_in[2];
    const float* shapedirs = (const float*)d_in[3];   // (V,3,10)  -> (20670,10)
    const float* posedirs  = (const float*)d_in[4];   // (V,3,207) -> (20670,207)
    const float* weights   = (const float*)d_in[5];   // (V,24)
    const float* Jreg      = (const float*)d_in[6];   // (24,V)
    const float* Jspin     = (const float*)d_in[7];   // (21,V)
    const float* Jextra    = (const float*)d_in[8];   // (9,V)
    const int*   parent    = (const int*)d_in[9];
    const int*   jmap      = (const int*)d_in[10];

    const int B = SMPL_B, N3 = SMPL_B * 3;

    // workspace layout (floats)
    float* ws     = (float*)d_ws;
    float* vbuf   = ws;                                  // B*V*3 = 10,583,040
    float* lrotT  = vbuf  + (size_t)B * SMPL_M;          // 207*B = 105,984
    float* Jbuf   = lrotT + (size_t)207 * B;             // B*72  = 36,864
    float* Gc     = Jbuf  + (size_t)B * 72;              // B*288 = 147,456
    float* jcat   = Gc    + (size_t)B * 288;             // B*162 = 82,944
    float* pelvis = jcat  + (size_t)B * 162;             // B*3

    float* out_v      = (float*)d_out;                   // (B,V,3)
    float* out_joints = out_v + (size_t)B * SMPL_M;      // (B,49,3)
    float* out_pelvis = out_joints + (size_t)B * 147;    // (B,1,3)

    const int mtiles32 = (SMPL_M + 31) / 32;             // 646 row-strips
    const dim3 gdirs((mtiles32 + 3) / 4, B / (16 * TN)); // (162, 8)

    // 1. v_shaped = v_template + shapedirs . beta^T     (WMMA, K=10)
    k_wmma_dirs<1, 0><<<gdirs, 128, 0, stream>>>(
        shapedirs, 10, beta, v_tmpl, vbuf, SMPL_M, B);

    // 2. J = J_regressor . v_shaped                     (WMMA, K=6890)
    k_wmma_reg<<<dim3(1, N3 / (16 * TN)), 64, 0, stream>>>(
        Jreg, 24, Jreg, 0, Jreg, 24, vbuf, Jbuf, N3);

    // 3. Rodrigues + lrotmin^T + kinematic chain + Gc
    k_pose<<<dim3((B + 31) / 32), 32, 0, stream>>>(
        pose, Jbuf, parent, lrotT, Gc, B);

    // 4. v_posed = v_shaped + posedirs . lrotmin^T      (WMMA, K=207, in place)
    k_wmma_dirs<0, 1><<<gdirs, 128, 0, stream>>>(
        posedirs, 207, lrotT, nullptr, vbuf, SMPL_M, B);

    // 5. skinning (in place on vbuf)
    k_skin<<<dim3((SMPL_V + 255) / 256, B), 256, 0, stream>>>(
        weights, Gc, vbuf);

    // 6. jcat = [Jreg; Jspin; Jextra] . v               (WMMA, K=6890)
    k_wmma_reg<<<dim3(1, N3 / (16 * TN)), 64, 0, stream>>>(
        Jreg, 24, Jspin, 21, Jextra, 54, vbuf, jcat, N3);

    // 7. joint map + pelvis
    k_map<<<dim3((B + 255) / 256), 256, 0, stream>>>(
        jcat, jmap, out_joints, out_pelvis, pelvis, B);

    // 8. out_v = v - pelvis
    int total = B * SMPL_M;
    k_sub<<<dim3((total + 255) / 256), 256, 0, stream>>>(
        vbuf, pelvis, out_v, total);

    (void)in_sizes; (void)n_in; (void)out_size; (void)ws_size;
}